// SoftCrossEntropyLoss_65231963292461
// MI455X (gfx1250) — compile-verified
//
#include <hip/hip_runtime.h>

typedef float f32x4 __attribute__((ext_vector_type(4)));
typedef float v2f   __attribute__((ext_vector_type(2)));
typedef float v8f   __attribute__((ext_vector_type(8)));

__device__ __forceinline__ float wave_max32(float x) {
#pragma unroll
    for (int off = 16; off >= 1; off >>= 1)
        x = fmaxf(x, __shfl_xor(x, off, 32));
    return x;
}

__device__ __forceinline__ float wave_sum32(float x) {
#pragma unroll
    for (int off = 16; off >= 1; off >>= 1)
        x += __shfl_xor(x, off, 32);
    return x;
}

// loss partial accumulator kernel: one wave (32 lanes) per row of K=128 floats.
// Each lane owns 4 contiguous columns (one float4 = global_load_b128).
__global__ __launch_bounds__(256) void sce_main_kernel(
    const float* __restrict__ Y_tp,
    const float* __restrict__ Y_t,
    double* __restrict__ accum,
    long long N)
{
    const int lane = threadIdx.x & 31;
    const int wib  = threadIdx.x >> 5;               // wave index in block (0..7)
    const long long wavesTotal = (long long)gridDim.x * (blockDim.x >> 5);
    const long long waveId     = (long long)blockIdx.x * (blockDim.x >> 5) + wib;

    const f32x4* __restrict__ tp4 = (const f32x4*)Y_tp;
    const f32x4* __restrict__ yt4 = (const f32x4*)Y_t;

    float acc = 0.0f;   // per-lane running sum of y * logp over assigned rows

    for (long long row = waveId; row < N; row += wavesTotal) {
        const long long base = row * 32 + lane;      // float4 index: 32 float4 per row
        // Stream both rows once, non-temporal (1 GiB total, read exactly once).
        f32x4 v = __builtin_nontemporal_load(tp4 + base);
        f32x4 y = __builtin_nontemporal_load(yt4 + base);

        // logsumexp over the row (wave32 cross-lane reductions)
        float m = fmaxf(fmaxf(v[0], v[1]), fmaxf(v[2], v[3]));
        m = wave_max32(m);
        float e = __expf(v[0] - m) + __expf(v[1] - m) +
                  __expf(v[2] - m) + __expf(v[3] - m);
        e = wave_sum32(e);
        const float lse = m + __logf(e);

        // sum_k y[k] * logp[k], this lane's 4 columns
        acc += y[0] * (v[0] - lse) + y[1] * (v[1] - lse) +
               y[2] * (v[2] - lse) + y[3] * (v[3] - lse);
    }

    // ---- Wave-level reduction of the 32 per-lane partials via WMMA f32 16x16x4.
    // A layout (32-bit 16x4): lanes 0-15 VGPR0 -> A[m,0], VGPR1 -> A[m,1];
    //                         lanes 16-31 VGPR0 -> A[m,2], VGPR1 -> A[m,3].
    // With A[.,0]=p[0..15], A[.,2]=p[16..31], others 0, and B = all-ones:
    //   D[m,n] = p[m] + p[m+16]  (independent of n).
    // D layout: lane L<16 holds D[0..7, L]; lane L>=16 holds D[8..15, L-16].
    // So sum of a lane's 8 D regs = half-total; xor-16 partner supplies the rest.
    v2f A; A[0] = acc;  A[1] = 0.0f;
    v2f B; B[0] = 1.0f; B[1] = 1.0f;
    v8f C = {};
    C = __builtin_amdgcn_wmma_f32_16x16x4_f32(
        /*neg_a=*/false, A, /*neg_b=*/false, B,
        /*c_mod=*/(short)0, C, /*reuse_a=*/false, /*reuse_b=*/false);
    float t = C[0] + C[1] + C[2] + C[3] + C[4] + C[5] + C[6] + C[7];
    t += __shfl_xor(t, 16, 32);   // now every lane holds the full wave sum

    __shared__ float waveSum[8];
    if (lane == 0) waveSum[wib] = t;
    __syncthreads();

    if (threadIdx.x == 0) {
        double s = 0.0;
        const int nw = blockDim.x >> 5;
        for (int i = 0; i < nw; ++i) s += (double)waveSum[i];
        atomicAdd(accum, s);       // global_atomic_add_f64, one per block
    }
}

__global__ void sce_init_kernel(double* accum) {
    if (threadIdx.x == 0 && blockIdx.x == 0) accum[0] = 0.0;
}

__global__ void sce_finalize_kernel(const double* __restrict__ accum,
                                    float* __restrict__ out) {
    if (threadIdx.x == 0 && blockIdx.x == 0) out[0] = (float)(-accum[0]);
}

extern "C" void kernel_launch(void* const* d_in, const int* in_sizes, int n_in,
                              void* d_out, int out_size, void* d_ws, size_t ws_size,
                              hipStream_t stream) {
    const float* Y_tp = (const float*)d_in[0];
    const float* Y_t  = (const float*)d_in[1];
    float* out = (float*)d_out;

    const long long total = (long long)in_sizes[0];
    const long long K = 128;
    const long long N = total / K;

    double* accum = (double*)d_ws;

    sce_init_kernel<<<1, 1, 0, stream>>>(accum);

    // 8192 blocks x 8 waves = 65,536 waves -> 16 rows/wave; ~16 MB of B128
    // loads in flight across the device, comfortably covering HBM latency
    // at 23.3 TB/s, while keeping same-address f64 atomics to 8192 total.
    const int blocks = 8192;
    sce_main_kernel<<<blocks, 256, 0, stream>>>(Y_tp, Y_t, accum, N);

    sce_finalize_kernel<<<1, 1, 0, stream>>>(accum, out);
}